// Structure_Information_54443005444380
// MI455X (gfx1250) — compile-verified
//
#include <hip/hip_runtime.h>
#include <hip/hip_bf16.h>

#define B_DIM 2
#define C_DIM 128
#define N_PTS 8192
#define K_NN  16
#define QK_STRIDE 132   // padded row stride (floats) to spread LDS banks
#define WV 2            // waves (points) per block in attention kernel

typedef __attribute__((ext_vector_type(2))) float v2f;
typedef __attribute__((ext_vector_type(4))) float v4f;
typedef __attribute__((ext_vector_type(8))) float v8f;

// ---------------------------------------------------------------------------
// Kernel A: F[m][b][n][c'] = sum_c W_m[c'][c] * feature[b][c][n]
// One wave per 16x16 output tile; V_WMMA_F32_16X16X4_F32, 32 K-steps.
// ---------------------------------------------------------------------------
__global__ __launch_bounds__(64) void proj_kernel(
    const float* __restrict__ feature,
    const float* __restrict__ Wq, const float* __restrict__ Wk,
    const float* __restrict__ Wv, float* __restrict__ F) {
  const int lane = threadIdx.x & 31;
  int tile = blockIdx.x * 2 + (threadIdx.x >> 5);
  const int NT = N_PTS / 16;                 // 512 n-tiles
  const int nt = tile % NT; tile /= NT;
  const int ct = tile % (C_DIM / 16); tile /= (C_DIM / 16);
  const int m  = tile % 3; tile /= 3;
  const int b  = tile;

  const float* W = (m == 0) ? Wq : ((m == 1) ? Wk : Wv);
  const float* feat = feature + (size_t)b * C_DIM * N_PTS;
  float* out = F + ((size_t)m * B_DIM + b) * N_PTS * C_DIM;

  const int r    = lane & 15;
  const int koff = (lane < 16) ? 0 : 2;
  const int c0   = ct * 16, n0 = nt * 16;

  const float* wrow = W + (size_t)(c0 + r) * C_DIM + koff;  // A: [M=c', K=c]
  const float* fcol = feat + n0 + r;                        // B: [K=c, N=n]

  v8f acc = {};
#pragma unroll 4
  for (int kk = 0; kk < C_DIM; kk += 4) {
    v2f a = *(const v2f*)(wrow + kk);
    v2f bb;
    bb.x = fcol[(size_t)(kk + koff) * N_PTS];
    bb.y = fcol[(size_t)(kk + koff + 1) * N_PTS];
    acc = __builtin_amdgcn_wmma_f32_16x16x4_f32(false, a, false, bb,
                                                (short)0, acc, false, false);
  }
  // D layout: lane<16 -> c' = c0+0..7, lane>=16 -> c' = c0+8..15, n = n0+r
  float* orow = out + (size_t)(n0 + r) * C_DIM + c0 + ((lane < 16) ? 0 : 8);
  v4f lo = {acc[0], acc[1], acc[2], acc[3]};
  v4f hi = {acc[4], acc[5], acc[6], acc[7]};
  *(v4f*)(orow)     = lo;
  *(v4f*)(orow + 4) = hi;
}

// ---------------------------------------------------------------------------
// Kernel B: per-point attention. One wave per point.
//   Q[k]=Fq[n]-Fq[idx_k]+bq staged in LDS (row means/norms reduced in-wave),
//   sim = Q*K^T via 32x V_WMMA_F32_16X16X4_F32, dual-branch diagonal softmax,
//   out row = sum_k w[k]*(Fv[idx_k]+bv), stored [B*N][C] (coalesced).
// ---------------------------------------------------------------------------
__global__ __launch_bounds__(WV * 32) void attn_kernel(
    const float* __restrict__ F, const long long* __restrict__ knn,
    const float* __restrict__ bq, const float* __restrict__ bk,
    const float* __restrict__ bv, float* __restrict__ wout) {
  __shared__ float smem[WV * (2 * 16 * QK_STRIDE + 80)];
  const int lane = threadIdx.x & 31;
  const int w = threadIdx.x >> 5;
  const int p = blockIdx.x * WV + w;
  const int b = p / N_PTS, n = p % N_PTS;

  float* Qs = smem + w * (2 * 16 * QK_STRIDE + 80);
  float* Ks = Qs + 16 * QK_STRIDE;
  float* st = Ks + 16 * QK_STRIDE;
  float* mqA = st;       float* qnA = st + 16;
  float* mkA = st + 32;  float* knA = st + 48;
  float* dwA = st + 64;

  const float* Fq = F + (size_t)b * N_PTS * C_DIM;
  const float* Fk = F + ((size_t)B_DIM + b) * N_PTS * C_DIM;
  const float* Fv = F + ((size_t)2 * B_DIM + b) * N_PTS * C_DIM;

  const int c4 = lane * 4;
  v4f cq  = *(const v4f*)(Fq + (size_t)n * C_DIM + c4);
  v4f ck  = *(const v4f*)(Fk + (size_t)n * C_DIM + c4);
  v4f bq4 = *(const v4f*)(bq + c4);
  v4f bk4 = *(const v4f*)(bk + c4);

  int myIdx = 0;
  if (lane < K_NN) myIdx = (int)knn[((size_t)b * N_PTS + n) * K_NN + lane];

  // ---- stage Q, K rows in LDS; reduce row mean & centered norm in-wave ----
  for (int k = 0; k < K_NN; ++k) {
    const int j = __shfl(myIdx, k, 32);
    v4f q = cq - *(const v4f*)(Fq + (size_t)j * C_DIM + c4) + bq4;
    *(v4f*)(Qs + k * QK_STRIDE + c4) = q;
    float s  = q.x + q.y + q.z + q.w;
    float ss = q.x * q.x + q.y * q.y + q.z * q.z + q.w * q.w;
    for (int d = 1; d < 32; d <<= 1) { s += __shfl_xor(s, d, 32); ss += __shfl_xor(ss, d, 32); }
    if (lane == 0) {
      float mq = s * (1.0f / (float)C_DIM);
      mqA[k] = mq;
      qnA[k] = sqrtf(fmaxf(ss - (float)C_DIM * mq * mq, 1e-12f));
    }
    v4f kf = ck - *(const v4f*)(Fk + (size_t)j * C_DIM + c4) + bk4;
    *(v4f*)(Ks + k * QK_STRIDE + c4) = kf;
    s  = kf.x + kf.y + kf.z + kf.w;
    ss = kf.x * kf.x + kf.y * kf.y + kf.z * kf.z + kf.w * kf.w;
    for (int d = 1; d < 32; d <<= 1) { s += __shfl_xor(s, d, 32); ss += __shfl_xor(ss, d, 32); }
    if (lane == 0) {
      float mk = s * (1.0f / (float)C_DIM);
      mkA[k] = mk;
      knA[k] = sqrtf(fmaxf(ss - (float)C_DIM * mk * mk, 1e-12f));
    }
  }
  __syncthreads();

  // ---- sim[k,q] = Q[k] . K[q] : M=16(k), N=16(q), K=128, f32 WMMA ----
  const int r    = lane & 15;
  const int koff = (lane < 16) ? 0 : 2;
  const float* qrow = Qs + r * QK_STRIDE + koff;   // A frag: Q[r][kk..]
  const float* krow = Ks + r * QK_STRIDE + koff;   // B frag: K^T[kk][r] = K[r][kk..]
  v8f acc = {};
#pragma unroll 4
  for (int kk = 0; kk < C_DIM; kk += 4) {
    v2f a  = *(const v2f*)(qrow + kk);
    v2f bb = *(const v2f*)(krow + kk);
    acc = __builtin_amdgcn_wmma_f32_16x16x4_f32(false, a, false, bb,
                                                (short)0, acc, false, false);
  }

  // ---- dual-branch row softmax, keep only diagonal weight ----
  const float mkq = mkA[r];
  const float knq = knA[r];
  const int diagBase = (lane < 16) ? 0 : 24;   // lane holding q==row in each half
#pragma unroll
  for (int jj = 0; jj < 8; ++jj) {
    const int row = jj + ((lane < 16) ? 0 : 8);
    const float s1  = acc[jj];
    const float mqr = mqA[row];
    const float qnr = qnA[row];
    // branch 1: plain dot-product softmax
    float m1 = s1;
    for (int d = 1; d < 16; d <<= 1) m1 = fmaxf(m1, __shfl_xor(m1, d, 16));
    float e1 = __expf(s1 - m1);
    float sum1 = e1;
    for (int d = 1; d < 16; d <<= 1) sum1 += __shfl_xor(sum1, d, 16);
    const float d1 = __shfl(e1, diagBase + jj, 32) / sum1;
    // branch 2: cosine-normalized, via sim - C*mq*mk identity
    float s2 = (s1 - (float)C_DIM * mqr * mkq) / (qnr * knq);
    float m2 = s2;
    for (int d = 1; d < 16; d <<= 1) m2 = fmaxf(m2, __shfl_xor(m2, d, 16));
    float e2 = __expf(s2 - m2);
    float sum2 = e2;
    for (int d = 1; d < 16; d <<= 1) sum2 += __shfl_xor(sum2, d, 16);
    const float d2 = __shfl(e2, diagBase + jj, 32) / sum2;
    if ((lane & 15) == 0) dwA[row] = 0.6f * d1 + 0.4f * d2;
  }
  __syncthreads();

  // ---- out[c] = sum_k dw[k] * (Fv[idx_k][c] + bv[c]) ----
  v4f bv4 = *(const v4f*)(bv + c4);
  v4f o = {};
  for (int k = 0; k < K_NN; ++k) {
    const int j = __shfl(myIdx, k, 32);
    const float wk = dwA[k];
    v4f v = *(const v4f*)(Fv + (size_t)j * C_DIM + c4) + bv4;
    o += v * wk;
  }
  *(v4f*)(wout + (size_t)p * C_DIM + c4) = o;
}

// ---------------------------------------------------------------------------
// Kernel C: per-channel batchnorm stats over (B,N); scale/shift precompute.
// ---------------------------------------------------------------------------
__global__ __launch_bounds__(256) void bnstats_kernel(
    const float* __restrict__ wbuf, const float* __restrict__ gamma,
    const float* __restrict__ beta, float* __restrict__ stats) {
  __shared__ float ssum[256], ssq[256];
  const int c = blockIdx.x, t = threadIdx.x;
  const int P = B_DIM * N_PTS;
  float s = 0.f, q = 0.f;
  for (int p = t; p < P; p += 256) {
    float v = wbuf[(size_t)p * C_DIM + c];
    s += v; q += v * v;
  }
  ssum[t] = s; ssq[t] = q;
  __syncthreads();
  for (int d = 128; d > 0; d >>= 1) {
    if (t < d) { ssum[t] += ssum[t + d]; ssq[t] += ssq[t + d]; }
    __syncthreads();
  }
  if (t == 0) {
    float mean = ssum[0] / (float)P;
    float var  = ssq[0] / (float)P - mean * mean;
    float sc   = gamma[c] * rsqrtf(var + 1e-5f);
    stats[c] = sc;
    stats[C_DIM + c] = beta[c] - mean * sc;
  }
}

// ---------------------------------------------------------------------------
// Kernel D: y[b][c][n] = relu(scale[c]*w[b][n][c] + shift[c])  (transpose)
// ---------------------------------------------------------------------------
__global__ __launch_bounds__(256) void bnapply_kernel(
    const float* __restrict__ wbuf, const float* __restrict__ stats,
    float* __restrict__ out) {
  const int idx = blockIdx.x * 256 + threadIdx.x;
  const int total = B_DIM * C_DIM * N_PTS;
  if (idx >= total) return;
  const int n = idx % N_PTS;
  const int c = (idx / N_PTS) % C_DIM;
  const int b = idx / (N_PTS * C_DIM);
  const float v = wbuf[((size_t)b * N_PTS + n) * C_DIM + c];
  const float y = stats[c] * v + stats[C_DIM + c];
  out[idx] = fmaxf(y, 0.0f);
}

// ---------------------------------------------------------------------------
extern "C" void kernel_launch(void* const* d_in, const int* in_sizes, int n_in,
                              void* d_out, int out_size, void* d_ws, size_t ws_size,
                              hipStream_t stream) {
  const float*     feature = (const float*)d_in[0];
  const long long* knn     = (const long long*)d_in[1];
  const float* Wq = (const float*)d_in[2];
  const float* bq = (const float*)d_in[3];
  const float* Wk = (const float*)d_in[4];
  const float* bk = (const float*)d_in[5];
  const float* Wv = (const float*)d_in[6];
  const float* bv = (const float*)d_in[7];
  const float* gamma = (const float*)d_in[8];
  const float* beta  = (const float*)d_in[9];

  float* ws = (float*)d_ws;
  const size_t BNC = (size_t)B_DIM * N_PTS * C_DIM;   // 2,097,152 floats
  float* Fbuf  = ws;                                  // 3*BNC: Fq,Fk,Fv
  float* wbuf  = ws + 3 * BNC;                        // BNC: pre-BN result
  float* stats = ws + 4 * BNC;                        // 256 floats

  // A: 2*3*8*512 = 24576 tiles, 2 waves/block -> 12288 blocks
  proj_kernel<<<12288, 64, 0, stream>>>(feature, Wq, Wk, Wv, Fbuf);
  // B: 16384 points, WV per block
  attn_kernel<<<(B_DIM * N_PTS) / WV, WV * 32, 0, stream>>>(Fbuf, knn, bq, bk, bv, wbuf);
  // C: one block per channel
  bnstats_kernel<<<C_DIM, 256, 0, stream>>>(wbuf, gamma, beta, stats);
  // D: elementwise, 2,097,152 / 256 = 8192 blocks
  bnapply_kernel<<<(B_DIM * C_DIM * N_PTS + 255) / 256, 256, 0, stream>>>(wbuf, stats, (float*)d_out);
}